// EAPNNForce_11261404250841
// MI455X (gfx1250) — compile-verified
//
#include <hip/hip_runtime.h>
#include <math.h>

// ---------------------------------------------------------------------------
// EAPNN featurization for MI455X (gfx1250, wave32, WMMA).
// GEMMs use V_WMMA_F32_16X16X4_F32; G_atom (16MB) is L2-resident so the
// af = (G_atom[i]+G_atom[j])/2 matrix is gathered on the fly (saves 160MB HBM).
// ---------------------------------------------------------------------------

typedef __attribute__((ext_vector_type(2))) float v2f;
typedef __attribute__((ext_vector_type(8))) float v8f;

#define PI_F 3.14159265358979323846f
#define RC_F 8.0f
#define NTYPE 10
#define MAXNB 16
#define ACSF_NMU 20
#define APSF_NMU 10
#define AF_K 200   // ACSF_NMU * NTYPE
#define PF_K 100   // APSF_NMU * NTYPE
#define AF_N 64
#define PF_N 32
#define FEATD 118  // 64 + 32 + 22

__device__ __forceinline__ void inv3x3(const float* b, float* inv) {
  float a00=b[0],a01=b[1],a02=b[2],a10=b[3],a11=b[4],a12=b[5],a20=b[6],a21=b[7],a22=b[8];
  float c00 =  (a11*a22 - a12*a21);
  float c01 = -(a10*a22 - a12*a20);
  float c02 =  (a10*a21 - a11*a20);
  float det = a00*c00 + a01*c01 + a02*c02;
  float id = 1.0f/det;
  inv[0]=c00*id;                 inv[1]=-(a01*a22-a02*a21)*id;  inv[2]= (a01*a12-a02*a11)*id;
  inv[3]=c01*id;                 inv[4]= (a00*a22-a02*a20)*id;  inv[5]=-(a00*a12-a02*a10)*id;
  inv[6]=c02*id;                 inv[7]=-(a00*a21-a01*a20)*id;  inv[8]= (a00*a11-a01*a10)*id;
}

__device__ __forceinline__ void pbc3(float dx, float dy, float dz,
                                     const float* box, const float* binv, float* o) {
  float s0 = dx*binv[0] + dy*binv[3] + dz*binv[6];
  float s1 = dx*binv[1] + dy*binv[4] + dz*binv[7];
  float s2 = dx*binv[2] + dy*binv[5] + dz*binv[8];
  s0 -= floorf(s0 + 0.5f); s1 -= floorf(s1 + 0.5f); s2 -= floorf(s2 + 0.5f);
  o[0] = s0*box[0] + s1*box[3] + s2*box[6];
  o[1] = s0*box[1] + s1*box[4] + s2*box[7];
  o[2] = s0*box[2] + s1*box[5] + s2*box[8];
}

__device__ __forceinline__ float softplusf(float x) {
  return (x > 30.0f) ? x : log1pf(__expf(x));
}

__device__ __forceinline__ float wave_max(float v) {
  #pragma unroll
  for (int o = 16; o > 0; o >>= 1) v = fmaxf(v, __shfl_xor(v, o, 32));
  return v;
}

// ---------------------------------------------------------------------------
__global__ void k_init(unsigned* maxes) {
  if (threadIdx.x < 2) maxes[threadIdx.x] = 0u;
}

// per (atom, neighbor): cache drn, f_cut, neighbor atype
__global__ void k_env(const float* __restrict__ pos, const float* __restrict__ box,
                      const int* __restrict__ nbl, const int* __restrict__ atype,
                      float* __restrict__ env_drn, float* __restrict__ env_fc,
                      int* __restrict__ env_at, int n_atoms) {
  int t = blockIdx.x*blockDim.x + threadIdx.x;
  if (t >= n_atoms*MAXNB) return;
  int a = t / MAXNB;
  int nb = nbl[t];
  if (nb < 0) { env_drn[t] = 0.0f; env_fc[t] = 0.0f; env_at[t] = 0; return; }
  float binv[9]; inv3x3(box, binv);
  float d[3];
  pbc3(pos[nb*3]-pos[a*3], pos[nb*3+1]-pos[a*3+1], pos[nb*3+2]-pos[a*3+2], box, binv, d);
  float drn = sqrtf(d[0]*d[0] + d[1]*d[1] + d[2]*d[2] + 1e-10f);
  float x = drn * (1.0f/RC_F);
  float fc = (x < 1.0f) ? 0.5f*(__cosf(PI_F*x) + 1.0f) : 0.0f;
  env_drn[t] = drn; env_fc[t] = fc; env_at[t] = atype[nb];
}

// per (atom, mu): G_atom[a][m][t] accumulation
__global__ void k_gatom(const float* __restrict__ acsf_mus, const float* __restrict__ acsf_eta,
                        const int* __restrict__ atype, const float* __restrict__ env_drn,
                        const float* __restrict__ env_fc, const int* __restrict__ env_at,
                        float* __restrict__ G_atom, int n_atoms) {
  int t = blockIdx.x*blockDim.x + threadIdx.x;
  if (t >= n_atoms*ACSF_NMU) return;
  int a = t / ACSF_NMU, m = t % ACSF_NMU;
  int at = atype[a];
  float eta = softplusf(acsf_eta[at]);
  float mu  = acsf_mus[at*ACSF_NMU + m];
  float acc[NTYPE] = {};
  #pragma unroll
  for (int n = 0; n < MAXNB; n++) {
    int e = a*MAXNB + n;
    float dd = env_drn[e] - mu;
    float g = __expf(-eta*dd*dd) * env_fc[e];
    int ta = env_at[e];
    #pragma unroll
    for (int q = 0; q < NTYPE; q++) acc[q] += (q == ta) ? g : 0.0f;
  }
  float* o = G_atom + (size_t)a*AF_K + m*NTYPE;
  #pragma unroll
  for (int q = 0; q < NTYPE; q++) o[q] = acc[q];
}

// per pair: dr_norm, buffer, one-hot feat cols, cos-gamma + mask/type cache
__global__ void k_pair(const float* __restrict__ pos, const float* __restrict__ box,
                       const int* __restrict__ pairs, const int* __restrict__ nbl,
                       const int* __restrict__ molID, const int* __restrict__ atype,
                       float* __restrict__ out_feats, float* __restrict__ out_drn,
                       float* __restrict__ out_buf, float* __restrict__ cosg,
                       int* __restrict__ pack, int P) {
  int p = blockIdx.x*blockDim.x + threadIdx.x;
  if (p >= P) return;
  int lo = pairs[p*3], hi = pairs[p*3+1];
  int sh = (hi - lo <= 0) ? 1 : 0;
  int iI = lo - sh, jI = hi - 2*sh;
  float bscale = (hi - lo > 0) ? 1.0f : 0.0f;
  float binv[9]; inv3x3(box, binv);
  float rij[3];
  pbc3(pos[jI*3]-pos[iI*3], pos[jI*3+1]-pos[iI*3+1], pos[jI*3+2]-pos[iI*3+2], box, binv, rij);
  float r2 = rij[0]*rij[0] + rij[1]*rij[1] + rij[2]*rij[2];
  float drn = sqrtf(r2);
  float binter = (molID[iI] == molID[jI]) ? 0.0f : 1.0f;
  float cut = (drn <= RC_F) ? 0.5f*(1.0f + __cosf(PI_F*drn*(1.0f/RC_F))) : 0.0f;
  float bnir = binter * bscale * cut;
  out_drn[p] = drn;
  out_buf[p] = bnir;
  float* frow = out_feats + (size_t)p*FEATD;
  #pragma unroll
  for (int c = 0; c < 22; c++) frow[96 + c] = 0.0f;
  frow[96  + atype[iI]] = 1.0f;
  frow[107 + atype[jI]] = 1.0f;
  float inr = rsqrtf(r2 + 1e-10f);
  float u0 = rij[0]*inr, u1 = rij[1]*inr, u2 = rij[2]*inr;
  size_t cb = (size_t)p * (2*MAXNB);
  #pragma unroll 4
  for (int n = 0; n < MAXNB; n++) {
    int jn = nbl[iI*MAXNB + n];
    bool mk = (jn != iI) && (jn != jI) && (jn != -1);
    float c = 0.0f; int pk = 0;
    if (mk) {
      float d[3];
      pbc3(pos[jn*3]-pos[iI*3], pos[jn*3+1]-pos[iI*3+1], pos[jn*3+2]-pos[iI*3+2], box, binv, d);
      float in2 = rsqrtf(d[0]*d[0]+d[1]*d[1]+d[2]*d[2] + 1e-10f);
      c = (d[0]*u0 + d[1]*u1 + d[2]*u2) * in2;
      pk = atype[jn] + 1;
    }
    cosg[cb + n] = c; pack[cb + n] = pk;
  }
  #pragma unroll 4
  for (int n = 0; n < MAXNB; n++) {
    int kn = nbl[jI*MAXNB + n];
    bool mk = (kn != iI) && (kn != jI) && (kn != -1);
    float c = 0.0f; int pk = 0;
    if (mk) {
      float d[3];
      pbc3(pos[kn*3]-pos[jI*3], pos[kn*3+1]-pos[jI*3+1], pos[kn*3+2]-pos[jI*3+2], box, binv, d);
      float in2 = rsqrtf(d[0]*d[0]+d[1]*d[1]+d[2]*d[2] + 1e-10f);
      c = -(d[0]*u0 + d[1]*u1 + d[2]*u2) * in2;
      pk = atype[kn] + 1;
    }
    cosg[cb + MAXNB + n] = c; pack[cb + MAXNB + n] = pk;
  }
}

// per (pair, mu): G_pair -> pf workspace + pf max-abs
__global__ void k_gpair(const float* __restrict__ apsf_mus, const float* __restrict__ apsf_eta,
                        const int* __restrict__ nbl, const int* __restrict__ pairs,
                        const int* __restrict__ atype, const float* __restrict__ cosg,
                        const int* __restrict__ pack, const float* __restrict__ out_buf,
                        float* __restrict__ pf, unsigned* __restrict__ max_pf,
                        int P, int n_atoms) {
  int t = blockIdx.x*blockDim.x + threadIdx.x;
  float lmax = 0.0f;
  if (t < P*APSF_NMU) {
    int p = t / APSF_NMU, m = t % APSF_NMU;
    int lo = pairs[p*3], hi = pairs[p*3+1];
    int sh = (hi - lo <= 0) ? 1 : 0;
    int iI = lo - sh, jI = hi - 2*sh;
    int j0 = nbl[iI*MAXNB]; if (j0 < 0) j0 += n_atoms;   // JAX negative-index wrap
    int k0 = nbl[jI*MAXNB]; if (k0 < 0) k0 += n_atoms;
    int ati = atype[j0], atj = atype[k0];
    float mui = apsf_mus[ati*APSF_NMU + m], muj = apsf_mus[atj*APSF_NMU + m];
    float ei = softplusf(apsf_eta[ati]), ej = softplusf(apsf_eta[atj]);
    float acc[NTYPE] = {};
    size_t cb = (size_t)p * (2*MAXNB);
    #pragma unroll 4
    for (int n = 0; n < MAXNB; n++) {
      int pk = pack[cb + n];
      if (pk) {
        float dd = cosg[cb + n] - mui;
        float g = __expf(-ei*dd*dd);
        #pragma unroll
        for (int q = 0; q < NTYPE; q++) acc[q] += (q == pk-1) ? g : 0.0f;
      }
    }
    #pragma unroll 4
    for (int n = 0; n < MAXNB; n++) {
      int pk = pack[cb + MAXNB + n];
      if (pk) {
        float dd = cosg[cb + MAXNB + n] - muj;
        float g = __expf(-ej*dd*dd);
        #pragma unroll
        for (int q = 0; q < NTYPE; q++) acc[q] += (q == pk-1) ? g : 0.0f;
      }
    }
    float s = 0.5f * out_buf[p];
    float* o = pf + (size_t)p*PF_K + m*NTYPE;
    #pragma unroll
    for (int q = 0; q < NTYPE; q++) {
      float v = acc[q] * s;
      o[q] = v;
      lmax = fmaxf(lmax, fabsf(v));
    }
  }
  lmax = wave_max(lmax);
  if ((threadIdx.x & 31) == 0) atomicMax(max_pf, __float_as_uint(lmax));
}

// per pair: max-abs of (G_atom[i]+G_atom[j])*0.5  (G_atom is L2-resident)
__global__ void k_afmax(const int* __restrict__ pairs, const float* __restrict__ G_atom,
                        unsigned* __restrict__ max_af, int P) {
  int p = blockIdx.x*blockDim.x + threadIdx.x;
  float lm = 0.0f;
  if (p < P) {
    int lo = pairs[p*3], hi = pairs[p*3+1];
    int sh = (hi - lo <= 0) ? 1 : 0;
    const float* gi = G_atom + (size_t)(lo - sh)*AF_K;
    const float* gj = G_atom + (size_t)(hi - 2*sh)*AF_K;
    #pragma unroll 8
    for (int k = 0; k < AF_K; k++)
      lm = fmaxf(lm, fabsf(0.5f*(gi[k] + gj[k])));
  }
  lm = wave_max(lm);
  if ((threadIdx.x & 31) == 0) atomicMax(max_af, __float_as_uint(lm));
}

// af GEMM: (P x 200) @ (200 x 64), A gathered from G_atom, V_WMMA_F32_16X16X4_F32
__global__ void k_gemm_af(const int* __restrict__ pairs, const float* __restrict__ G_atom,
                          const float* __restrict__ Wa, const float* __restrict__ ba,
                          const unsigned* __restrict__ max_af, float* __restrict__ out_feats,
                          int P) {
  int lane = threadIdx.x & 31;
  int tile = blockIdx.x*(blockDim.x >> 5) + (threadIdx.x >> 5);
  if (tile*16 >= P) return;                        // wave-uniform
  float scale = 0.5f / __uint_as_float(*max_af);   // includes the 0.5 averaging
  int row = lane & 15;
  int pr = min(tile*16 + row, P - 1);
  int lo = pairs[pr*3], hi = pairs[pr*3+1];
  int sh = (hi - lo <= 0) ? 1 : 0;
  size_t gi = (size_t)(lo - sh)*AF_K, gj = (size_t)(hi - 2*sh)*AF_K;
  int khalf = (lane >> 4) << 1;                    // lanes 16-31 hold K+2, K+3
  v8f c0 = {}, c1 = {}, c2 = {}, c3 = {};
  for (int k0 = 0; k0 < AF_K; k0 += 4) {
    int kA = k0 + khalf;
    v2f a;
    a.x = (G_atom[gi + kA]     + G_atom[gj + kA])     * scale;
    a.y = (G_atom[gi + kA + 1] + G_atom[gj + kA + 1]) * scale;
    const float* wb = Wa + (size_t)kA*AF_N + row;
    v2f b0, b1, b2, b3;
    b0.x = wb[0];  b0.y = wb[AF_N];
    b1.x = wb[16]; b1.y = wb[AF_N + 16];
    b2.x = wb[32]; b2.y = wb[AF_N + 32];
    b3.x = wb[48]; b3.y = wb[AF_N + 48];
    c0 = __builtin_amdgcn_wmma_f32_16x16x4_f32(false, a, false, b0, (short)0, c0, false, false);
    c1 = __builtin_amdgcn_wmma_f32_16x16x4_f32(false, a, false, b1, (short)0, c1, false, false);
    c2 = __builtin_amdgcn_wmma_f32_16x16x4_f32(false, a, false, b2, (short)0, c2, false, false);
    c3 = __builtin_amdgcn_wmma_f32_16x16x4_f32(false, a, false, b3, (short)0, c3, false, false);
  }
  int mOff = (lane >> 4) << 3;
  int nI = lane & 15;
  #pragma unroll
  for (int r = 0; r < 8; r++) {
    int pp = tile*16 + r + mOff;
    if (pp < P) {
      float* fr = out_feats + (size_t)pp*FEATD;
      fr[nI]      = fmaxf(c0[r] + ba[nI],      0.0f);
      fr[16 + nI] = fmaxf(c1[r] + ba[16 + nI], 0.0f);
      fr[32 + nI] = fmaxf(c2[r] + ba[32 + nI], 0.0f);
      fr[48 + nI] = fmaxf(c3[r] + ba[48 + nI], 0.0f);
    }
  }
}

// pf GEMM: (P x 100) @ (100 x 32)
__global__ void k_gemm_pf(const float* __restrict__ pf, const float* __restrict__ Wp,
                          const float* __restrict__ bp, const unsigned* __restrict__ max_pf,
                          float* __restrict__ out_feats, int P) {
  int lane = threadIdx.x & 31;
  int tile = blockIdx.x*(blockDim.x >> 5) + (threadIdx.x >> 5);
  if (tile*16 >= P) return;
  float scale = 1.0f / __uint_as_float(*max_pf);
  int row = lane & 15;
  int pr = min(tile*16 + row, P - 1);
  const float* arow = pf + (size_t)pr*PF_K;
  int khalf = (lane >> 4) << 1;
  v8f c0 = {}, c1 = {};
  for (int k0 = 0; k0 < PF_K; k0 += 4) {
    int kA = k0 + khalf;
    v2f a;
    a.x = arow[kA]     * scale;
    a.y = arow[kA + 1] * scale;
    const float* wb = Wp + (size_t)kA*PF_N + row;
    v2f b0, b1;
    b0.x = wb[0];  b0.y = wb[PF_N];
    b1.x = wb[16]; b1.y = wb[PF_N + 16];
    c0 = __builtin_amdgcn_wmma_f32_16x16x4_f32(false, a, false, b0, (short)0, c0, false, false);
    c1 = __builtin_amdgcn_wmma_f32_16x16x4_f32(false, a, false, b1, (short)0, c1, false, false);
  }
  int mOff = (lane >> 4) << 3;
  int nI = lane & 15;
  #pragma unroll
  for (int r = 0; r < 8; r++) {
    int pp = tile*16 + r + mOff;
    if (pp < P) {
      float* fr = out_feats + (size_t)pp*FEATD;
      fr[64 + nI] = fmaxf(c0[r] + bp[nI],      0.0f);
      fr[80 + nI] = fmaxf(c1[r] + bp[16 + nI], 0.0f);
    }
  }
}

// ---------------------------------------------------------------------------
extern "C" void kernel_launch(void* const* d_in, const int* in_sizes, int n_in,
                              void* d_out, int out_size, void* d_ws, size_t ws_size,
                              hipStream_t stream) {
  const float* pos      = (const float*)d_in[0];
  const float* box      = (const float*)d_in[1];
  const float* acsf_mus = (const float*)d_in[2];
  const float* acsf_eta = (const float*)d_in[3];
  const float* apsf_mus = (const float*)d_in[4];
  const float* apsf_eta = (const float*)d_in[5];
  const float* Wa       = (const float*)d_in[6];
  const float* ba       = (const float*)d_in[7];
  const float* Wp       = (const float*)d_in[8];
  const float* bp       = (const float*)d_in[9];
  const int*   pairs    = (const int*)d_in[10];
  const int*   nbl      = (const int*)d_in[11];
  // d_in[12] topo_mask: redundant (== nbl != -1), unused
  const int*   molID    = (const int*)d_in[13];
  const int*   atype    = (const int*)d_in[14];

  int n_atoms = in_sizes[0] / 3;
  int P       = in_sizes[10] / 3;

  float* out      = (float*)d_out;
  float* out_feat = out;
  float* out_drn  = out + (size_t)P * FEATD;
  float* out_buf  = out_drn + P;

  // workspace carve-out (256B aligned)
  char* ws = (char*)d_ws;
  size_t off = 0;
  auto take = [&](size_t bytes) -> char* {
    char* p = ws + off;
    off += (bytes + 255) & ~(size_t)255;
    return p;
  };
  float* env_drn = (float*)take((size_t)n_atoms * MAXNB * 4);
  float* env_fc  = (float*)take((size_t)n_atoms * MAXNB * 4);
  int*   env_at  = (int*)  take((size_t)n_atoms * MAXNB * 4);
  float* G_atom  = (float*)take((size_t)n_atoms * AF_K * 4);
  float* cosg    = (float*)take((size_t)P * 2 * MAXNB * 4);
  int*   pack    = (int*)  take((size_t)P * 2 * MAXNB * 4);
  float* pfw     = (float*)take((size_t)P * PF_K * 4);
  unsigned* maxes = (unsigned*)take(8);   // [0]=max_af, [1]=max_pf

  const int B = 256;
  k_init<<<1, 64, 0, stream>>>(maxes);

  int nEnv = n_atoms * MAXNB;
  k_env<<<(nEnv + B - 1)/B, B, 0, stream>>>(pos, box, nbl, atype,
                                            env_drn, env_fc, env_at, n_atoms);

  int nGA = n_atoms * ACSF_NMU;
  k_gatom<<<(nGA + B - 1)/B, B, 0, stream>>>(acsf_mus, acsf_eta, atype,
                                             env_drn, env_fc, env_at, G_atom, n_atoms);

  k_pair<<<(P + B - 1)/B, B, 0, stream>>>(pos, box, pairs, nbl, molID, atype,
                                          out_feat, out_drn, out_buf, cosg, pack, P);

  int nGP = P * APSF_NMU;
  k_gpair<<<(nGP + B - 1)/B, B, 0, stream>>>(apsf_mus, apsf_eta, nbl, pairs, atype,
                                             cosg, pack, out_buf, pfw, maxes + 1,
                                             P, n_atoms);

  k_afmax<<<(P + B - 1)/B, B, 0, stream>>>(pairs, G_atom, maxes, P);

  int tiles = (P + 15) / 16;
  int wavesPerBlock = 4;                      // 128 threads = 4 wave32
  int gemmBlocks = (tiles + wavesPerBlock - 1) / wavesPerBlock;
  k_gemm_af<<<gemmBlocks, wavesPerBlock*32, 0, stream>>>(pairs, G_atom, Wa, ba,
                                                         maxes, out_feat, P);
  k_gemm_pf<<<gemmBlocks, wavesPerBlock*32, 0, stream>>>(pfw, Wp, bp,
                                                         maxes + 1, out_feat, P);
}